// MemoryLoadAttention_62380105007343
// MI455X (gfx1250) — compile-verified
//
#include <hip/hip_runtime.h>
#include <math.h>

// ---------------- problem constants (match reference) ----------------
#define DIMV   64
#define BATCH  32
#define LLEN   16384
#define SCALEF 1.25f       // 10/sqrt(64)
#define SLOPE  0.01f

// ---------------- decomposition ----------------
#define SPLIT  32                          // chunks per batch -> 1024 blocks
#define CHUNK  (LLEN / SPLIT)              // 512 keys per block
#define WAVES  8                           // 256 threads, wave32
#define TILE   16                          // keys per WMMA tile
#define TILES_PER_WAVE (CHUNK / (WAVES * TILE))   // 4
#define PART_STRIDE 66                     // m, s, acc[64]

typedef __attribute__((ext_vector_type(2))) float v2f;
typedef __attribute__((ext_vector_type(8))) float v8f;

__device__ __forceinline__ float wave_max32(float v) {
#pragma unroll
  for (int off = 16; off >= 1; off >>= 1)
    v = fmaxf(v, __shfl_xor(v, off, 32));
  return v;
}
__device__ __forceinline__ float wave_sum32(float v) {
#pragma unroll
  for (int off = 16; off >= 1; off >>= 1)
    v += __shfl_xor(v, off, 32);
  return v;
}

// One block = one (batch, L-chunk). Streams K/V once, online softmax,
// QK dots via V_WMMA_F32_16X16X4_F32 chained over 16 K-chunks (D=64).
// Outer tile loop is kept rolled (#pragma unroll 1) to stay under 256 VGPRs:
// latency hiding comes from wave occupancy, not single-wave tile pipelines.
__global__ __launch_bounds__(256)
void attn_partial_kernel(const float* __restrict__ q,
                         const float* __restrict__ k,
                         const float* __restrict__ v,
                         const int*   __restrict__ qpos_p,
                         const int*   __restrict__ kpos,
                         float*       __restrict__ part)
{
  const int split = blockIdx.x;
  const int b     = blockIdx.y;
  const int tid   = threadIdx.x;
  const int wave  = tid >> 5;
  const int lane  = tid & 31;
  const int hi    = lane >> 4;      // 0: K-rows {0,1}, 1: K-rows {2,3} of each chunk
  const int keysub = lane & 15;     // key index within tile owned by this lane

  const float qpos = (float)qpos_p[0];

  // Preload query chunks for the WMMA A operand (A[m][k] = q[4c+k], all rows equal,
  // so the per-lane M mapping is irrelevant; only the K<->half-wave split matters).
  const float* qb = q + (size_t)b * DIMV;
  float2 qa[16];
#pragma unroll
  for (int c = 0; c < 16; ++c)
    qa[c] = *(const float2*)(qb + 4 * c + 2 * hi);

  const float* kb = k    + (size_t)b * LLEN * DIMV;
  const float* vb = v    + (size_t)b * LLEN * DIMV;
  const int*   pb = kpos + (size_t)b * LLEN;

  float m = -INFINITY, s = 0.0f, acc0 = 0.0f, acc1 = 0.0f;

  const int chunk_start = split * CHUNK;

#pragma unroll 1
  for (int t = 0; t < TILES_PER_WAVE; ++t) {
    const int tile_base = chunk_start + (wave + t * WAVES) * TILE;
    const float* krow = kb + (size_t)tile_base * DIMV;
    const float* vrow = vb + (size_t)tile_base * DIMV;

    // Prefetch next tile (global_prefetch_b8); 32 lanes x 128B covers the 4KB tile.
    if (t + 1 < TILES_PER_WAVE) {
      const float* nk = krow + (size_t)(WAVES * TILE) * DIMV;
      const float* nv = vrow + (size_t)(WAVES * TILE) * DIMV;
      __builtin_prefetch(nk + lane * 32, 0, 1);
      __builtin_prefetch(nv + lane * 32, 0, 1);
    }

    // ---- load the full 16x64 K tile for this wave (16 independent b64 loads) ----
    float2 kk[16];
#pragma unroll
    for (int c = 0; c < 16; ++c)
      kk[c] = *(const float2*)(krow + (size_t)keysub * DIMV + 4 * c + 2 * hi);

    // ---- scores for 16 keys via chained f32 WMMA (16 x K=4 chunks = D 64) ----
    // B[k][n] = K[key n][4c+k]; lane n<16 supplies K-rows {0,1}, lane n+16 rows {2,3}.
    v8f cacc = {};
#pragma unroll
    for (int c = 0; c < 16; ++c) {
      v2f a  = {qa[c].x, qa[c].y};
      v2f bm = {kk[c].x, kk[c].y};
      cacc = __builtin_amdgcn_wmma_f32_16x16x4_f32(
          /*neg_a=*/false, a, /*neg_b=*/false, bm,
          /*c_mod=*/(short)0, cacc, /*reuse_a=*/false, /*reuse_b=*/false);
    }
    // D[m][n] identical across m (A rows equal) => lane holds score of key (lane&15).
    float score = cacc[0] * SCALEF;
    const float dist = fabsf(qpos - (float)pb[tile_base + keysub]);
    score -= SLOPE * dist;

    // ---- online softmax update (lanes j and j+16 duplicate key j) ----
    const float tmax    = wave_max32(score);
    const float mnew    = fmaxf(m, tmax);
    const float rescale = __expf(m - mnew);          // exp(-inf)=0 on first tile
    const float p       = __expf(score - mnew);
    s = s * rescale + 0.5f * wave_sum32(p);          // halve: each key counted twice
    acc0 *= rescale;
    acc1 *= rescale;

    // ---- V accumulation: lane owns dims {2*lane, 2*lane+1}; rows coalesced ----
#pragma unroll
    for (int j = 0; j < 16; ++j) {
      const float  wj = __shfl(p, j, 32);
      const float2 vv = *(const float2*)(vrow + (size_t)j * DIMV + 2 * lane);
      acc0 = fmaf(wj, vv.x, acc0);
      acc1 = fmaf(wj, vv.y, acc1);
    }
    m = mnew;
  }

  // ---- merge the 8 waves of this block, emit one partial (m, s, acc[64]) ----
  __shared__ float sm[WAVES], ss[WAVES], sacc[WAVES][DIMV];
  if (lane == 0) { sm[wave] = m; ss[wave] = s; }
  sacc[wave][2 * lane]     = acc0;
  sacc[wave][2 * lane + 1] = acc1;
  __syncthreads();

  if (tid < DIMV) {
    float M = -INFINITY;
#pragma unroll
    for (int w = 0; w < WAVES; ++w) M = fmaxf(M, sm[w]);
    float S = 0.0f, A = 0.0f;
#pragma unroll
    for (int w = 0; w < WAVES; ++w) {
      const float e = __expf(sm[w] - M);
      S += ss[w] * e;
      A += sacc[w][tid] * e;
    }
    float* outp = part + ((size_t)b * SPLIT + split) * PART_STRIDE;
    if (tid == 0) { outp[0] = M; outp[1] = S; }
    outp[2 + tid] = A;
  }
}

// Merge the SPLIT partials of each batch: log-sum-exp combine, normalize.
__global__ __launch_bounds__(DIMV)
void attn_final_kernel(const float* __restrict__ part, float* __restrict__ out)
{
  const int b = blockIdx.x;
  const int d = threadIdx.x;  // 0..63
  const float* pb = part + (size_t)b * SPLIT * PART_STRIDE;

  float M = -INFINITY;
#pragma unroll 4
  for (int p = 0; p < SPLIT; ++p) M = fmaxf(M, pb[p * PART_STRIDE]);

  float S = 0.0f, A = 0.0f;
#pragma unroll 4
  for (int p = 0; p < SPLIT; ++p) {
    const float e = __expf(pb[p * PART_STRIDE] - M);
    S += pb[p * PART_STRIDE + 1] * e;
    A += pb[p * PART_STRIDE + 2 + d] * e;
  }
  out[(size_t)b * DIMV + d] = A / S;
}

extern "C" void kernel_launch(void* const* d_in, const int* in_sizes, int n_in,
                              void* d_out, int out_size, void* d_ws, size_t ws_size,
                              hipStream_t stream)
{
  const float* q    = (const float*)d_in[0];  // [B, 64]
  const float* k    = (const float*)d_in[1];  // [B, L, 64]
  const float* v    = (const float*)d_in[2];  // [B, L, 64]
  const int*   qpos = (const int*)  d_in[3];  // scalar
  const int*   kpos = (const int*)  d_in[4];  // [B, L]
  float*       part = (float*)d_ws;           // B*SPLIT*66 floats (~270 KB)
  float*       out  = (float*)d_out;          // [B, 64]

  (void)in_sizes; (void)n_in; (void)out_size; (void)ws_size;

  dim3 grid(SPLIT, BATCH);
  attn_partial_kernel<<<grid, WAVES * 32, 0, stream>>>(q, k, v, qpos, kpos, part);
  attn_final_kernel<<<BATCH, DIMV, 0, stream>>>(part, out);
}